// composed_loss_function_57724360458572
// MI455X (gfx1250) — compile-verified
//
#include <hip/hip_runtime.h>
#include <hip/hip_bf16.h>
#include <math.h>

typedef __attribute__((ext_vector_type(2))) float v2f;
typedef __attribute__((ext_vector_type(8))) float v8f;

#define MU_PRIOR 0.6447f
#define NBLK 1024      // partial rows; 1024*16*4B = 64KB of d_ws
#define NTHR 256       // 8 wave32 per block

// ---------------------------------------------------------------------------
// Pass 1: grid-stride streaming reduction. Each thread owns ~N/(NBLK*NTHR)
// rows (~4), accumulates 5 fp32 sums, reduces wave->block, writes a 16-wide
// partial row (cols 0..4 = S1..S5, cols 5..15 = 0) laid out for WMMA-B use.
// ---------------------------------------------------------------------------
__global__ __launch_bounds__(NTHR) void loss_partials(
    const float* __restrict__ pred_, const float* __restrict__ yobs,
    const float* __restrict__ rrs,   const float* __restrict__ rrsp,
    const float* __restrict__ nanarr,const float* __restrict__ covz,
    const float* __restrict__ muz,   float* __restrict__ partials, int N)
{
    // rrs_cov_inv = inv(diag(se^2)) -> diagonal weights 1/se^2
    const float w[5] = {
        1.0f / (0.0015f  * 0.0015f ),
        1.0f / (0.0012f  * 0.0012f ),
        1.0f / (0.001f   * 0.001f  ),
        1.0f / (0.00086f * 0.00086f),
        1.0f / (0.00057f * 0.00057f)
    };

    float a0 = 0.f, a1 = 0.f, a2 = 0.f, a3 = 0.f, a4 = 0.f;

    for (int i = blockIdx.x * NTHR + threadIdx.x; i < N; i += NBLK * NTHR) {
        // ---- rrs weighted quadratic (diagonal metric) ----
        {
            const float* r  = rrs  + i * 5;
            const float* rp = rrsp + i * 5;
            float s = 0.f;
            #pragma unroll
            for (int j = 0; j < 5; ++j) {
                float d = r[j] - rp[j];
                s = fmaf(w[j] * d, d, s);
            }
            a0 += s;
        }
        // ---- obs quad / lens (Y_cov_inv = I; lens = #non-NaN per row) ----
        {
            const float* p = pred_  + i * 9;
            const float* y = yobs   + i * 9;
            const float* z = nanarr + i * 9;
            float q = 0.f, lens = 0.f;
            #pragma unroll
            for (int j = 0; j < 9; ++j) {
                float d = p[j] - y[j];
                q = fmaf(d, d, q);
                float x = z[j];
                lens += (x == x) ? 1.f : 0.f;   // !isnan
            }
            a1 += q / lens;                     // lens==0 -> inf, same as ref
        }
        // ---- 3x3 logdet + trace (s_a_inv = I/4.9 so tr term = trace/4.9) ----
        {
            const float* c = covz + i * 9;
            float c00=c[0], c01=c[1], c02=c[2];
            float c10=c[3], c11=c[4], c12=c[5];
            float c20=c[6], c21=c[7], c22=c[8];
            float det = c00 * (c11*c22 - c12*c21)
                      - c01 * (c10*c22 - c12*c20)
                      + c02 * (c10*c21 - c11*c20);
            a2 += logf(det);
            a3 += c00 + c11 + c22;
        }
        // ---- ||mu_z - mu_prior||^2 ----
        {
            const float* m = muz + i * 3;
            #pragma unroll
            for (int j = 0; j < 3; ++j) {
                float d = m[j] - MU_PRIOR;
                a4 = fmaf(d, d, a4);
            }
        }
    }

    // ---- intra-wave reduce (wave32) ----
    #pragma unroll
    for (int off = 16; off > 0; off >>= 1) {
        a0 += __shfl_xor(a0, off, 32);
        a1 += __shfl_xor(a1, off, 32);
        a2 += __shfl_xor(a2, off, 32);
        a3 += __shfl_xor(a3, off, 32);
        a4 += __shfl_xor(a4, off, 32);
    }

    // ---- cross-wave reduce via LDS (8 waves) ----
    __shared__ float red[NTHR / 32][5];
    const int wid  = threadIdx.x >> 5;
    const int lane = threadIdx.x & 31;
    if (lane == 0) {
        red[wid][0] = a0; red[wid][1] = a1; red[wid][2] = a2;
        red[wid][3] = a3; red[wid][4] = a4;
    }
    __syncthreads();
    if (threadIdx.x < 16) {
        const int col = threadIdx.x;
        float v = 0.f;
        if (col < 5) {
            #pragma unroll
            for (int wv = 0; wv < NTHR / 32; ++wv) v += red[wv][col];
        }
        partials[blockIdx.x * 16 + col] = v;   // 16-wide row, zero-padded
    }
}

// ---------------------------------------------------------------------------
// Pass 2: one wave32. Column-sum the NBLK x 16 partial matrix with a chain of
// V_WMMA_F32_16X16X4_F32 ops: D = ones(16x4) x B(4x16) + C, so every D row
// accumulates per-column sums. B layout (lanes 0-15 = N, lane-half = K-half,
// VGPR0/1 = K parity) mirrors the ISA's 16x4 32-bit A table; A = all-ones is
// layout-invariant. Then lane 0 applies the closed-form loss.
// ---------------------------------------------------------------------------
__global__ void loss_finalize(const float* __restrict__ partials,
                              const float* __restrict__ params, int nparams,
                              float* __restrict__ out, int N)
{
    const int lane = threadIdx.x;          // 0..31, EXEC all ones
    const int n    = lane & 15;            // column index
    const int kh   = lane >> 4;            // K half

    v2f a; a.x = 1.0f; a.y = 1.0f;         // ones A-matrix
    v8f c = {};                            // C/D accumulator

    for (int base = 0; base < NBLK; base += 4) {
        v2f b;
        b.x = partials[(base + 2 * kh + 0) * 16 + n];  // K = 2*kh
        b.y = partials[(base + 2 * kh + 1) * 16 + n];  // K = 2*kh+1
        c = __builtin_amdgcn_wmma_f32_16x16x4_f32(
                /*neg_a=*/false, a, /*neg_b=*/false, b,
                /*c_mod=*/(short)0, c, /*reuse_a=*/false, /*reuse_b=*/false);
    }

    // D[0][n] (VGPR0, lanes 0-15) = total of accumulator column n
    float col = c[0];
    float S1 = __shfl(col, 0, 32);   // rrs weighted quad
    float S2 = __shfl(col, 1, 32);   // obs quad / lens
    float S3 = __shfl(col, 2, 32);   // sum logdet(cov_z)
    float S4 = __shfl(col, 3, 32);   // sum trace(cov_z)
    float S5 = __shfl(col, 4, 32);   // sum ||mu_z - mu||^2

    if (lane == 0) {
        const float nf = (float)N;
        float rrs_error = S1 / (5.0f * nf);
        float obs_error = S2 / nf;
        float logdet_sa = logf(4.9f * 4.9f * 4.9f);
        float DK = 0.5f * (nf * logdet_sa - S3 + S4 * (1.0f / 4.9f)) / nf
                 + 0.5f * S5 / (4.9f * 3.0f * nf);
        float l2 = 0.f;
        for (int j = 0; j < nparams; ++j) {
            float d = params[j] - 1.0f;
            l2 = fmaf(d, d, l2);
        }
        l2 /= (float)nparams;
        out[0] = rrs_error + 10.0f * obs_error + DK + l2;
    }
}

extern "C" void kernel_launch(void* const* d_in, const int* in_sizes, int n_in,
                              void* d_out, int out_size, void* d_ws, size_t ws_size,
                              hipStream_t stream)
{
    const float* pred_  = (const float*)d_in[0];  // [N,9]
    const float* yobs   = (const float*)d_in[1];  // [N,9]
    const float* rrs    = (const float*)d_in[2];  // [N,5]
    const float* rrsp   = (const float*)d_in[3];  // [N,5]
    const float* nanarr = (const float*)d_in[4];  // [N,9]
    const float* covz   = (const float*)d_in[5];  // [N,3,3]
    const float* muz    = (const float*)d_in[6];  // [N,3]
    const float* params = (const float*)d_in[7];  // [64]

    const int N       = in_sizes[0] / 9;
    const int nparams = in_sizes[7];

    float* partials = (float*)d_ws;               // NBLK*16 floats = 64 KB

    loss_partials<<<NBLK, NTHR, 0, stream>>>(pred_, yobs, rrs, rrsp, nanarr,
                                             covz, muz, partials, N);
    loss_finalize<<<1, 32, 0, stream>>>(partials, params, nparams,
                                        (float*)d_out, N);
}